// ShapeBottleneckModel_37529424233206
// MI455X (gfx1250) — compile-verified
//
#include <hip/hip_runtime.h>
#include <hip/hip_bf16.h>
#include <math.h>

// ---------------------------------------------------------------------------
// ShapeBottleneck on MI455X (gfx1250, wave32)
//   Kernel 1: per-(batch, bank) block — z-normalize into LDS, stage shapelet
//             bank in LDS, sliding mean-L1 with lanes striding over windows,
//             wave min-reduction. Uses exp(-d^2) monotonicity: max_p=exp(-dmin^2).
//   Kernel 2: head GEMM out = probs @ W_out^T via V_WMMA_F32_16X16X4_F32,
//             both operands staged in LDS (B zero-padded 10->16 rows) so the
//             K-loop is branch-free ds_load_b64 + wmma with full EXEC.
// ---------------------------------------------------------------------------

#define T_LEN 336
#define NCH   7
#define NSH   5
#define NPAIR (NSH * NCH)   // 35
#define FEAT  140           // 4 banks * 35
#define MAXL  168

__global__ __launch_bounds__(256) void sbm_shapelet_kernel(
    const float* __restrict__ x,   // (32,336,7)
    const float* __restrict__ w0,  // (5,7,34)
    const float* __restrict__ w1,  // (5,7,68)
    const float* __restrict__ w2,  // (5,7,101)
    const float* __restrict__ w3,  // (5,7,168)
    float* __restrict__ dists_out, // (32,140)
    float* __restrict__ probs_out) // (32,140)
{
    __shared__ float s_xn[NCH][T_LEN];          // 9408 B
    __shared__ float s_w[NSH * NCH * MAXL];     // 23520 B

    const int b    = blockIdx.x >> 2;
    const int bank = blockIdx.x & 3;

    int l;
    const float* wsrc;
    switch (bank) {
        case 0:  l = 34;  wsrc = w0; break;
        case 1:  l = 68;  wsrc = w1; break;
        case 2:  l = 101; wsrc = w2; break;
        default: l = 168; wsrc = w3; break;
    }
    const int tcount = T_LEN - l + 1;

    const int tid  = threadIdx.x;
    const int lane = tid & 31;
    const int wave = tid >> 5;

    // Stage this bank's shapelets in LDS (contiguous (n,c,l) layout).
    const int wtot = NSH * NCH * l;
    for (int i = tid; i < wtot; i += 256) s_w[i] = wsrc[i];

    // Per-channel z-normalization (ddof=1), one wave per channel.
    if (wave < NCH) {
        const int m = wave;
        float sum = 0.f, sumsq = 0.f;
        for (int t = lane; t < T_LEN; t += 32) {
            float v = x[(b * T_LEN + t) * NCH + m];
            sum   += v;
            sumsq += v * v;
        }
        #pragma unroll
        for (int off = 16; off > 0; off >>= 1) {
            sum   += __shfl_xor(sum,   off, 32);
            sumsq += __shfl_xor(sumsq, off, 32);
        }
        const float mu  = sum * (1.0f / T_LEN);
        float var = (sumsq - sum * mu) * (1.0f / (T_LEN - 1));
        var = fmaxf(var, 0.0f);
        const float inv = 1.0f / (sqrtf(var) + 1e-8f);
        for (int t = lane; t < T_LEN; t += 32) {
            float v = x[(b * T_LEN + t) * NCH + m];
            s_xn[m][t] = (v - mu) * inv;
        }
    }
    __syncthreads();

    // One wave per (n,m) pair; lanes stride over window positions t.
    const float inv_l = 1.0f / (float)l;
    for (int pair = wave; pair < NPAIR; pair += 8) {
        const int n = pair / NCH;
        const int m = pair % NCH;
        const float* __restrict__ wp = &s_w[(n * NCH + m) * l];
        const float* __restrict__ xp = &s_xn[m][0];

        float dmin = 3.4e38f;
        for (int t = lane; t < tcount; t += 32) {
            float s = 0.f;
            #pragma unroll 4
            for (int j = 0; j < l; ++j)
                s += fabsf(xp[t + j] - wp[j]);
            dmin = fminf(dmin, s * inv_l);
        }
        #pragma unroll
        for (int off = 16; off > 0; off >>= 1)
            dmin = fminf(dmin, __shfl_xor(dmin, off, 32));

        if (lane == 0) {
            const int col = bank * NPAIR + pair;       // concat order: bank-major
            dists_out[b * FEAT + col] = dmin;
            probs_out[b * FEAT + col] = expf(-dmin * dmin);  // EPS_GATE = 1
        }
    }
}

// ---------------------------------------------------------------------------
// Head: out(32,10) = probs(32,140) @ W_out(10,140)^T  via f32 WMMA 16x16x4.
// Per ISA 7.12.2 (wave32):
//   A 16x4 : lane L -> row M=L%16, half=L/16; VGPR j holds A[M][2*half + j]
//   B 4x16 : lane L -> col N=L%16;            VGPR j holds B[2*half + j][N]
//   C/D    : VGPR v holds row (v + 8*half), col N=L%16
// Both operands staged in LDS; B padded to 16 rows with zeros so the WMMA
// loop has no divergence (EXEC all-ones throughout, as the ISA requires).
// ---------------------------------------------------------------------------

typedef float v2f __attribute__((ext_vector_type(2)));
typedef float v8f __attribute__((ext_vector_type(8)));

__global__ __launch_bounds__(64) void sbm_head_kernel(
    const float* __restrict__ probs, // (32,140)
    const float* __restrict__ Wout,  // (10,140)
    float* __restrict__ out,         // (32,10)
    float* __restrict__ loss)        // scalar
{
    __shared__ float s_a[32 * FEAT];   // probs, row-major           (17920 B)
    __shared__ float s_b[16 * FEAT];   // Wout zero-padded to 16 rows (8960 B)

    const int tid  = threadIdx.x;
    const int lane = tid & 31;
    const int wave = tid >> 5;

    // Stage A (32x140) and padded B (16x140; rows 10..15 = 0) into LDS.
    for (int i = tid; i < 32 * FEAT; i += 64) s_a[i] = probs[i];
    for (int i = tid; i < 16 * FEAT; i += 64) s_b[i] = (i < 10 * FEAT) ? Wout[i] : 0.0f;
    __syncthreads();

    const int mo   = wave * 16;       // M-tile origin: 0 or 16
    const int half = lane >> 4;
    const int nm   = lane & 15;       // M (for A) / N (for B,D) within tile
    const int M    = mo + nm;

    v8f c = {};
    #pragma unroll 5
    for (int kk = 0; kk < FEAT; kk += 4) {
        const int k = kk + 2 * half;          // 8-byte aligned (k even)
        v2f a  = *(const v2f*)&s_a[M * FEAT + k];
        v2f bm = *(const v2f*)&s_b[nm * FEAT + k];
        c = __builtin_amdgcn_wmma_f32_16x16x4_f32(
                /*neg_a=*/false, a, /*neg_b=*/false, bm,
                /*c_mod=*/(short)0, c, /*reuse_a=*/false, /*reuse_b=*/false);
    }

    #pragma unroll
    for (int v = 0; v < 8; ++v) {
        const int row = mo + v + 8 * half;
        if (nm < 10) out[row * 10 + nm] = c[v];
    }

    // loss = 0.1 * mean(|W_out|), one-wave reduction, plain overwrite store.
    if (wave == 0) {
        float s = 0.f;
        for (int i = lane; i < 10 * FEAT; i += 32) s += fabsf(Wout[i]);
        #pragma unroll
        for (int off = 16; off > 0; off >>= 1) s += __shfl_xor(s, off, 32);
        if (lane == 0) *loss = 0.1f * (s * (1.0f / (10.0f * FEAT)));
    }
}

extern "C" void kernel_launch(void* const* d_in, const int* in_sizes, int n_in,
                              void* d_out, int out_size, void* d_ws, size_t ws_size,
                              hipStream_t stream) {
    (void)in_sizes; (void)n_in; (void)d_ws; (void)ws_size; (void)out_size;

    const float* x    = (const float*)d_in[0];
    const float* w0   = (const float*)d_in[1];
    const float* w1   = (const float*)d_in[2];
    const float* w2   = (const float*)d_in[3];
    const float* w3   = (const float*)d_in[4];
    const float* Wout = (const float*)d_in[5];

    float* out_p   = (float*)d_out;             // (32,10)   = 320
    float* dists_p = out_p + 320;               // (32,140)  = 4480
    float* probs_p = out_p + 320 + 4480;        // (32,140)  = 4480
    float* loss_p  = out_p + 320 + 4480 + 4480; // scalar

    sbm_shapelet_kernel<<<128, 256, 0, stream>>>(x, w0, w1, w2, w3,
                                                 dists_p, probs_p);
    sbm_head_kernel<<<1, 64, 0, stream>>>(probs_p, Wout, out_p, loss_p);
}